// SpectralConv3d_65592740545050
// MI455X (gfx1250) — compile-verified
//
#include <hip/hip_runtime.h>

// ---------------------------------------------------------------------------
// Hartley-FNO spectral conv for MI455X (gfx1250, wave32).
// dht(x)[k] = ReF3(x)[k]*(1 - sin(2pi*(k1+k2+k3)/64)) + ImF3(x)[k]*cos(...)
// All dense transform stages on V_WMMA_F32_16X16X4_F32.
// Basis matrices are staged in LDS in exact WMMA operand order:
//   A-side: row-major (adjacent K pair -> one ds_load_b64)
//   B-side: row-pair interleaved [k/2][n]{rowk,rowk+1} -> one ds_load_b64
// Global mode buffers (A_re/A_im, outm) are stored pair-interleaved over
// their contraction axis so K2/K4 B fetches are single global_load_b64.
// ---------------------------------------------------------------------------

typedef float v2f __attribute__((ext_vector_type(2)));
typedef float v8f __attribute__((ext_vector_type(8)));

#define ERR_T 0.1f

__device__ __forceinline__ v8f wmma4(v2f a, v2f b, v8f c) {
  // D(16x16,f32) = A(16x4,f32) * B(4x16,f32) + C
  return __builtin_amdgcn_wmma_f32_16x16x4_f32(
      false, a, false, b, (short)0, c, false, false);
}
__device__ __forceinline__ v2f ld2(const float* p) {
  return *(const v2f*)__builtin_assume_aligned(p, 8);
}

__device__ __forceinline__ void trig64(float* tc, float* ts, float* tsn, int t) {
  if (t < 64) {  // threads 0..63 = waves 0,1 fully -> no partial-wave EXEC
    float s, c;
    sincosf(6.283185307179586f * (float)t * (1.0f / 64.0f), &s, &c);
    tc[t] = c; ts[t] = s; tsn[t] = -s;
  }
}

// dst[r*cols + c] = tab[(r*c)&63]   (A-side, row-major)
__device__ __forceinline__ void fill_rm(float* dst, const float* tab, int rows,
                                        int cols, int t) {
  const int total = rows * cols;
  for (int i = t; i < total; i += 256) {
    const int r = i / cols, c = i - r * cols;
    dst[i] = tab[(r * c) & 63];
  }
}
// B-side row-pair layout: float2 dst[kp][c] = {tab[(2kp)c], tab[(2kp+1)c]}
__device__ __forceinline__ void fill_pr(float* dst, const float* tab, int khalf,
                                        int cols, int t) {
  const int total = khalf * cols;
  for (int i = t; i < total; i += 256) {
    const int kp = i / cols, c = i - kp * cols;
    dst[2 * i]     = tab[((2 * kp) * c) & 63];
    dst[2 * i + 1] = tab[((2 * kp + 1) * c) & 63];
  }
}

// Forward z/y transform of one 64x64 (n2,n3) slab in LDS.
// T1p: pair-layout scratch, re plane [0..1023], im plane [1024..2047].
// pZc/pZs: stage-Z B pair tables (cos / -sin), 32x16 pairs.
// Yc/Ys/Ysn: stage-Y A tables, row-major 16x64 (cos / sin / -sin).
// Output element (k2,k3) -> gA[(k2*16+k3)*2] (n1-pair-interleaved global).
__device__ __forceinline__ void fwd_zy(const float* xs, float* T1p,
                                       const float* pZc, const float* pZs,
                                       const float* Yc, const float* Ys,
                                       const float* Ysn, float* gAre,
                                       float* gAim, int wave, int lane) {
  const int mA = lane & 15;
  const int kk = (lane >> 4) << 1;   // K offset 0|2
  const int kh = kk >> 1;            // pair-row offset 0|1
  const int dOff = (lane >> 4) << 3; // D row offset 0|8

  { // Stage Z: T1[n2][k3] = sum_n3 xs * e^{-i 2pi k3 n3/64}; waves0-3 Re,4-7 Im
    const int mt = (wave & 3) << 4;
    const int isIm = wave >> 2;                       // scalar (wave uniform)
    const float* arow = xs + (mt + mA) * 64 + kk;
    const float* brow = (isIm ? pZs : pZc) + mA * 2 + kh * 32;
    v8f acc = {};
    for (int k0 = 0; k0 < 64; k0 += 4)
      acc = wmma4(ld2(arow + k0), ld2(brow + (k0 >> 1) * 32), acc);
    float* dst = T1p + isIm * 1024;
#pragma unroll
    for (int r = 0; r < 8; ++r) {
      const int row = mt + dOff + r;
      dst[(row >> 1) * 32 + mA * 2 + (row & 1)] = acc[r];
    }
  }
  __syncthreads();
  if (wave < 2) { // Stage Y: A[k2][k3] = sum_n2 e^{-i 2pi k2 n2/64} * T1
    const int isIm = wave;
    // Re = cosA*T1r + sinA*T1i ; Im = cosA*T1i - sinA*T1r
    const float* A0 = Yc + mA * 64 + kk;
    const float* A1 = (isIm ? Ysn : Ys) + mA * 64 + kk;
    const float* B0 = T1p + (isIm ? 1024 : 0) + mA * 2 + kh * 32;
    const float* B1 = T1p + (isIm ? 0 : 1024) + mA * 2 + kh * 32;
    v8f acc = {};
    for (int k0 = 0; k0 < 64; k0 += 4)
      acc = wmma4(ld2(A0 + k0), ld2(B0 + (k0 >> 1) * 32), acc);
    for (int k0 = 0; k0 < 64; k0 += 4)
      acc = wmma4(ld2(A1 + k0), ld2(B1 + (k0 >> 1) * 32), acc);
    float* dst = isIm ? gAim : gAre;
#pragma unroll
    for (int r = 0; r < 8; ++r)
      dst[((dOff + r) * 16 + mA) * 2] = acc[r];
  }
}

// K1: x slab -> partial complex modes A[bc][n1-pairs][k2*16+k3]
__global__ __launch_bounds__(256) void k_fwd(const float* __restrict__ x,
                                             float* __restrict__ A_re,
                                             float* __restrict__ A_im) {
  __shared__ __align__(16) float xs[4096];
  __shared__ __align__(16) float T1p[2048];
  __shared__ __align__(16) float pZc[1024], pZs[1024];
  __shared__ __align__(16) float Yc[1024], Ys[1024], Ysn[1024];
  __shared__ float tc[64], ts[64], tsn[64];
  const int t = threadIdx.x;
  trig64(tc, ts, tsn, t);
  __syncthreads();
  fill_pr(pZc, tc, 32, 16, t);
  fill_pr(pZs, tsn, 32, 16, t);
  fill_rm(Yc, tc, 16, 64, t);
  fill_rm(Ys, ts, 16, 64, t);
  fill_rm(Ysn, tsn, 16, 64, t);
  const int slab = blockIdx.x;  // B*C*64
  const float4* xp4 = reinterpret_cast<const float4*>(x + (long)slab * 4096);
  float4* xs4 = reinterpret_cast<float4*>(xs);
  for (int i = t; i < 1024; i += 256) xs4[i] = xp4[i];
  __syncthreads();
  const int bc = slab >> 6, n1 = slab & 63;
  const long aoff = (long)bc * 16384 + (n1 >> 1) * 512 + (n1 & 1);
  const int wave = __builtin_amdgcn_readfirstlane(t >> 5);
  fwd_zy(xs, T1p, pZc, pZs, Yc, Ys, Ysn, A_re + aoff, A_im + aoff, wave,
         t & 31);
}

// K2: contract n1 -> k1, apply Hartley phase combine -> real modes Xt
__global__ __launch_bounds__(256) void k_fwd_x(const float* __restrict__ A_re,
                                               const float* __restrict__ A_im,
                                               float* __restrict__ Xt) {
  __shared__ __align__(16) float Xc[1024], Xs[1024], Xsn[1024]; // [k1=16][n1=64]
  __shared__ float tc[64], ts[64], tsn[64];
  const int t = threadIdx.x;
  trig64(tc, ts, tsn, t);
  __syncthreads();
  fill_rm(Xc, tc, 16, 64, t);
  fill_rm(Xs, ts, 16, 64, t);
  fill_rm(Xsn, tsn, 16, 64, t);
  __syncthreads();
  const long bc = blockIdx.x;  // B*C
  const float* Ar = A_re + bc * 16384;  // pair-interleaved over n1
  const float* Ai = A_im + bc * 16384;
  float* X = Xt + bc * 4096;
  const int wave = __builtin_amdgcn_readfirstlane(t >> 5);
  const int lane = t & 31;
  const int mA = lane & 15, kk = (lane >> 4) << 1, kh = kk >> 1;
  const int dOff = (lane >> 4) << 3;
  const float* ac_ = Xc + mA * 64 + kk;
  const float* as_ = Xs + mA * 64 + kk;
  const float* an_ = Xsn + mA * 64 + kk;
  for (int nt = wave; nt < 16; nt += 8) {
    const int n0 = nt * 16;
    const float* br_ = Ar + (n0 + mA) * 2 + kh * 512;
    const float* bi_ = Ai + (n0 + mA) * 2 + kh * 512;
    v8f accR = {}, accI = {};
    for (int k0 = 0; k0 < 64; k0 += 4) {
      const v2f br = ld2(br_ + (k0 >> 1) * 512);
      const v2f bi = ld2(bi_ + (k0 >> 1) * 512);
      accR = wmma4(ld2(ac_ + k0), br, accR);  // Fr = cos*Ar + sin*Ai
      accR = wmma4(ld2(as_ + k0), bi, accR);
      accI = wmma4(ld2(ac_ + k0), bi, accI);  // Fi = cos*Ai - sin*Ar
      accI = wmma4(ld2(an_ + k0), br, accI);
    }
#pragma unroll
    for (int r = 0; r < 8; ++r) {
      const int k1 = dOff + r, col = n0 + mA;
      const int ph = (k1 + (col >> 4) + (col & 15)) & 63;
      X[k1 * 256 + col] = accR[r] * (1.0f - ts[ph]) + accI[r] * tc[ph];
    }
  }
}

// K3: per-mode channel mix; outm stored pair-interleaved over k1 for K4.
__global__ __launch_bounds__(256) void k_conv(const float* __restrict__ Xt,
                                              const float* __restrict__ w,
                                              float* __restrict__ outm,
                                              int accumulate) {
  const int idx = blockIdx.x * 256 + threadIdx.x;  // B*C*4096
  const int k = idx & 4095;
  const int o = (idx >> 12) & 31;
  const int b = idx >> 17;
  const int k1 = k >> 8, col = k & 255;
  const int paddr = (idx & ~4095) + ((k1 >> 1) << 9) + (col << 1) + (k1 & 1);
  const float* xb = Xt + b * 131072 + k;
  const float* wp = w + o * 4096 + k;
  float s = accumulate ? outm[paddr] : 0.0f;
#pragma unroll 8
  for (int i = 0; i < 32; ++i) s += xb[i * 4096] * wp[i * 131072];
  outm[paddr] = s;
}

// K4: expand k1 -> n1 (complex): G1[n1][col] = sum_{k1<16} e^{-i..} outm
__global__ __launch_bounds__(256) void k_inv_x(const float* __restrict__ outm,
                                               float* __restrict__ G1r,
                                               float* __restrict__ G1i) {
  __shared__ __align__(16) float Ic[1024], Isn[1024];  // [n1=64][k1=16]
  __shared__ float tc[64], ts[64], tsn[64];
  const int t = threadIdx.x;
  trig64(tc, ts, tsn, t);
  __syncthreads();
  fill_rm(Ic, tc, 64, 16, t);
  fill_rm(Isn, tsn, 64, 16, t);
  __syncthreads();
  const long bc = blockIdx.x;
  const float* O = outm + bc * 4096;  // pair-interleaved over k1
  float* Gr = G1r + bc * 16384;
  float* Gi = G1i + bc * 16384;
  const int wave = __builtin_amdgcn_readfirstlane(t >> 5);
  const int lane = t & 31;
  const int mA = lane & 15, kk = (lane >> 4) << 1, kh = kk >> 1;
  const int dOff = (lane >> 4) << 3;
  for (int p = wave; p < 64; p += 8) {
    const int m0 = (p >> 4) << 4, n0 = (p & 15) << 4;
    const float* ac_ = Ic + (m0 + mA) * 16 + kk;
    const float* an_ = Isn + (m0 + mA) * 16 + kk;
    const float* b_ = O + (n0 + mA) * 2 + kh * 512;
    v8f accR = {}, accI = {};
    for (int k0 = 0; k0 < 16; k0 += 4) {
      const v2f b = ld2(b_ + (k0 >> 1) * 512);
      accR = wmma4(ld2(ac_ + k0), b, accR);
      accI = wmma4(ld2(an_ + k0), b, accI);
    }
#pragma unroll
    for (int r = 0; r < 8; ++r) {
      Gr[(m0 + dOff + r) * 256 + n0 + mA] = accR[r];
      Gi[(m0 + dOff + r) * 256 + n0 + mA] = accI[r];
    }
  }
}

// K5: expand y,z + phase combine + 1/64^3.
// MODE 0: write final output. MODE 1: mask residual, re-run fwd z/y in-place.
template <int MODE>
__global__ __launch_bounds__(256) void k_inv_yz(
    const float* __restrict__ G1r, const float* __restrict__ G1i,
    const float* __restrict__ x, float* __restrict__ out,
    float* __restrict__ A_re, float* __restrict__ A_im) {
  __shared__ float tc[64], ts[64], tsn[64];
  __shared__ __align__(16) float Grp[256], Gip[256];           // pair over k2
  __shared__ __align__(16) float Yc2[1024], Ys2[1024], Ysn2[1024]; // [n2][k2]
  __shared__ __align__(16) float pc2[1024], ps2[1024], psn2[1024]; // pair[k3/2][n3]
  __shared__ __align__(16) float Tr[1024], Ti[1024];           // [n2=64][k3=16]
  __shared__ __align__(16) float xs[4096];
  __shared__ __align__(16) float T1p[2048];
  __shared__ __align__(16) float pZc[1024], pZs[1024];
  __shared__ __align__(16) float Yc[1024], Ys[1024], Ysn[1024];
  const int t = threadIdx.x;
  trig64(tc, ts, tsn, t);
  __syncthreads();
  fill_rm(Yc2, tc, 64, 16, t);
  fill_rm(Ys2, ts, 64, 16, t);
  fill_rm(Ysn2, tsn, 64, 16, t);
  fill_pr(pc2, tc, 8, 64, t);
  fill_pr(ps2, ts, 8, 64, t);
  fill_pr(psn2, tsn, 8, 64, t);
  if (MODE == 1) {
    fill_pr(pZc, tc, 32, 16, t);
    fill_pr(pZs, tsn, 32, 16, t);
    fill_rm(Yc, tc, 16, 64, t);
    fill_rm(Ys, ts, 16, 64, t);
    fill_rm(Ysn, tsn, 16, 64, t);
  }
  const int slab = blockIdx.x;  // B*C*64
  const int n1 = slab & 63;
  { // load G1 slab, store into k2-pair layout
    const float gr = G1r[(long)slab * 256 + t];
    const float gi = G1i[(long)slab * 256 + t];
    const int k2 = t >> 4, k3 = t & 15;
    const int pi = (k2 >> 1) * 32 + k3 * 2 + (k2 & 1);
    Grp[pi] = gr;
    Gip[pi] = gi;
  }
  __syncthreads();
  const int wave = __builtin_amdgcn_readfirstlane(t >> 5);
  const int lane = t & 31;
  const int mA = lane & 15, kk = (lane >> 4) << 1, kh = kk >> 1;
  const int dOff = (lane >> 4) << 3;
  { // Stage Y: T[n2][k3]; Tr = cos*Gr + sin*Gi ; Ti = cos*Gi - sin*Gr
    const int mt = (wave & 3) << 4;
    const int isIm = wave >> 2;
    const float* A0 = Yc2 + (mt + mA) * 16 + kk;
    const float* A1 = (isIm ? Ysn2 : Ys2) + (mt + mA) * 16 + kk;
    const float* B0 = (isIm ? Gip : Grp) + mA * 2 + kh * 32;
    const float* B1 = (isIm ? Grp : Gip) + mA * 2 + kh * 32;
    v8f acc = {};
    for (int k0 = 0; k0 < 16; k0 += 4)
      acc = wmma4(ld2(A0 + k0), ld2(B0 + (k0 >> 1) * 32), acc);
    for (int k0 = 0; k0 < 16; k0 += 4)
      acc = wmma4(ld2(A1 + k0), ld2(B1 + (k0 >> 1) * 32), acc);
    float* dst = isIm ? Ti : Tr;
#pragma unroll
    for (int r = 0; r < 8; ++r) dst[(mt + dOff + r) * 16 + mA] = acc[r];
  }
  __syncthreads();
  // Stage Z + combine: Fr = Tr*cos + Ti*sin ; Fi = Ti*cos - Tr*sin
  const float invn = 1.0f / 262144.0f;
  for (int p = wave; p < 16; p += 8) {
    const int m0 = (p >> 2) << 4, n0 = (p & 3) << 4;
    const float* ar_ = Tr + (m0 + mA) * 16 + kk;
    const float* ai_ = Ti + (m0 + mA) * 16 + kk;
    const float* bc_ = pc2 + (n0 + mA) * 2 + kh * 128;
    const float* bs_ = ps2 + (n0 + mA) * 2 + kh * 128;
    const float* bn_ = psn2 + (n0 + mA) * 2 + kh * 128;
    v8f accR = {}, accI = {};
    for (int k0 = 0; k0 < 16; k0 += 4) {
      const v2f ar = ld2(ar_ + k0), ai = ld2(ai_ + k0);
      const v2f bcv = ld2(bc_ + (k0 >> 1) * 128);
      const v2f bsv = ld2(bs_ + (k0 >> 1) * 128);
      const v2f bnv = ld2(bn_ + (k0 >> 1) * 128);
      accR = wmma4(ar, bcv, accR);
      accR = wmma4(ai, bsv, accR);
      accI = wmma4(ai, bcv, accI);
      accI = wmma4(ar, bnv, accI);
    }
#pragma unroll
    for (int r = 0; r < 8; ++r) {
      const int n2 = m0 + dOff + r, n3 = n0 + mA;
      const int ph = (n1 + n2 + n3) & 63;
      const float v = (accR[r] * (1.0f - ts[ph]) + accI[r] * tc[ph]) * invn;
      if (MODE == 0) {
        out[(long)slab * 4096 + n2 * 64 + n3] = v;
      } else {
        const float xv = x[(long)slab * 4096 + n2 * 64 + n3];
        xs[n2 * 64 + n3] = (fabsf(xv - v) > ERR_T) ? xv : 0.0f;
      }
    }
  }
  if (MODE == 1) {
    __syncthreads();  // xs complete
    const int bc = slab >> 6;
    const long aoff = (long)bc * 16384 + (n1 >> 1) * 512 + (n1 & 1);
    fwd_zy(xs, T1p, pZc, pZs, Yc, Ys, Ysn, A_re + aoff, A_im + aoff, wave,
           lane);
  }
}

extern "C" void kernel_launch(void* const* d_in, const int* in_sizes, int n_in,
                              void* d_out, int out_size, void* d_ws,
                              size_t ws_size, hipStream_t stream) {
  const float* x   = (const float*)d_in[0];  // (4,32,64,64,64)
  const float* w1  = (const float*)d_in[1];  // (32,32,16,16,16)
  const float* rw1 = (const float*)d_in[2];  // (32,32,16,16,16)
  float* out = (float*)d_out;

  // Workspace (floats): A (2x 2,097,152), G1 (2x 2,097,152),
  // Xt (524,288), outm (524,288) => 36 MiB.
  float* A_re = (float*)d_ws;
  float* A_im = A_re + 2097152;
  float* G1r  = A_im + 2097152;
  float* G1i  = G1r + 2097152;
  float* Xt   = G1i + 2097152;
  float* outm = Xt + 524288;

  const dim3 blk(256);
  // Pass 1
  k_fwd   <<<8192, blk, 0, stream>>>(x, A_re, A_im);
  k_fwd_x <<<128,  blk, 0, stream>>>(A_re, A_im, Xt);
  k_conv  <<<2048, blk, 0, stream>>>(Xt, w1, outm, 0);
  k_inv_x <<<128,  blk, 0, stream>>>(outm, G1r, G1i);
  // Reconstruction + mask fused with second forward transform
  k_inv_yz<1><<<8192, blk, 0, stream>>>(G1r, G1i, x, nullptr, A_re, A_im);
  // Pass 2 (refine)
  k_fwd_x <<<128,  blk, 0, stream>>>(A_re, A_im, Xt);
  k_conv  <<<2048, blk, 0, stream>>>(Xt, rw1, outm, 1);
  k_inv_x <<<128,  blk, 0, stream>>>(outm, G1r, G1i);
  k_inv_yz<0><<<8192, blk, 0, stream>>>(G1r, G1i, x, out, nullptr, nullptr);
}